// VectorQuantizer_48773648613460
// MI455X (gfx1250) — compile-verified
//
#include <hip/hip_runtime.h>
#include <stdint.h>

#define B_ROWS 65536
#define C_CODES 1024
#define D_DIM 64
#define BM 64                 // rows per workgroup
#define NBLK (B_ROWS / BM)    // 1024 workgroups

typedef float v2f __attribute__((ext_vector_type(2)));
typedef float v4f __attribute__((ext_vector_type(4)));
typedef float v8f __attribute__((ext_vector_type(8)));

// One workgroup: 64 rows x 1024 codes, 8 wave32 = 4 M-tiles x 2 column halves.
// q tile (64x1024 f32 = 256 KB) staged in LDS, normalized in place, streamed out once.
__global__ __launch_bounds__(256, 1)
void vq_main(const float* __restrict__ z, const float* __restrict__ cb,
             float* __restrict__ q_out, float* __restrict__ zq_out,
             float* __restrict__ loss_ws)
{
    extern __shared__ __align__(16) char smem[];
    float* qS   = (float*)smem;                                   // 64*1024
    float* zS   = qS + BM * C_CODES;                              // 64*64
    float* zn2S = zS + BM * D_DIM;                                // 64
    float* rsS  = zn2S + BM;                                      // 64 row sums
    unsigned long long* keyS = (unsigned long long*)(rsS + BM);   // 64 argmax keys
    float* lossS = (float*)(keyS + BM);                           // 1

    const int tid = threadIdx.x;
    const int wg  = blockIdx.x;

    // ---------------- phase 0: stage z tile, init reductions ----------------
    if (tid < BM) { rsS[tid] = 0.0f; keyS[tid] = 0ull; }
    if (tid == 0) lossS[0] = 0.0f;
    {
        const v4f* zg  = (const v4f*)(z + (size_t)wg * BM * D_DIM);
        v4f*       zs4 = (v4f*)zS;
        #pragma unroll
        for (int i = 0; i < 4; ++i) zs4[tid + i * 256] = zg[tid + i * 256];
    }
    __syncthreads();
    if (tid < BM) {
        float s = 0.0f;
        const float* zr = zS + tid * D_DIM;
        #pragma unroll
        for (int d = 0; d < D_DIM; ++d) s += zr[d] * zr[d];
        zn2S[tid] = s;
    }
    __syncthreads();

    // ---------------- phase 1: fp32 WMMA GEMM + fused q/argmax ----------------
    const int lane  = tid & 31;
    const int wave  = tid >> 5;
    const int half  = lane >> 4;   // K half / M half per WMMA layout
    const int lm    = lane & 15;
    const int mtile = wave >> 1;   // 0..3 : which 16-row tile
    const int chalf = wave & 1;    // 0..1 : which 512-column half

    // A fragments: z[mtile*16+lm][K], K = 2*half + j + 4*kc  (resident all 64 K)
    v2f aF[16];
    {
        const v2f* zr = (const v2f*)(zS + (mtile * 16 + lm) * D_DIM + 2 * half);
        #pragma unroll
        for (int kc = 0; kc < 16; ++kc) aF[kc] = zr[2 * kc];
    }
    float myzn2[8];
    #pragma unroll
    for (int i = 0; i < 8; ++i) myzn2[i] = zn2S[mtile * 16 + 8 * half + i];

    float rs[8];
    unsigned long long key[8];
    #pragma unroll
    for (int i = 0; i < 8; ++i) { rs[i] = 0.0f; key[i] = 0ull; }

    for (int t = 0; t < 32; ++t) {
        const int col0 = chalf * 512 + t * 16;
        const int n    = col0 + lm;                       // this lane's output column
        const v2f* crow = (const v2f*)(cb + (size_t)n * D_DIM + 2 * half);

        v8f acc = {};
        float w2h = 0.0f;                                 // half of ||e_n||^2
        #pragma unroll
        for (int kc = 0; kc < 16; ++kc) {
            v2f bF = crow[2 * kc];                        // codebook fragment (L2-resident)
            w2h += bF[0] * bF[0] + bF[1] * bF[1];
            acc = __builtin_amdgcn_wmma_f32_16x16x4_f32(
                      false, aF[kc], false, bF, (short)0, acc, false, false);
        }
        const float wn2 = w2h + __shfl_xor(w2h, 16);      // other K-half lives in lane^16

        #pragma unroll
        for (int i = 0; i < 8; ++i) {
            const int   row = mtile * 16 + 8 * half + i;  // C/D layout: M = i + 8*half
            const float d   = myzn2[i] + wn2 - 2.0f * acc[i];
            const float qv  = 1.0f / (1.0f + d);          // (1/(1+d))^((1+1)/2)
            qS[row * C_CODES + n] = qv;
            rs[i] += qv;
            unsigned ub = __float_as_uint(d);             // monotonic float->uint
            ub = (ub & 0x80000000u) ? ~ub : (ub | 0x80000000u);
            unsigned long long k =
                ((unsigned long long)ub << 32) | (unsigned)(~(unsigned)n); // ties -> lowest col
            key[i] = (k > key[i]) ? k : key[i];
        }
    }

    // butterfly reduce across the 16 lanes of each half, then combine via LDS atomics
    #pragma unroll
    for (int i = 0; i < 8; ++i) {
        #pragma unroll
        for (int off = 8; off > 0; off >>= 1) {
            rs[i] += __shfl_xor(rs[i], off);
            unsigned hi  = (unsigned)(key[i] >> 32);
            unsigned lo  = (unsigned)(key[i] & 0xFFFFFFFFull);
            unsigned ohi = __shfl_xor(hi, off);
            unsigned olo = __shfl_xor(lo, off);
            unsigned long long ok = ((unsigned long long)ohi << 32) | olo;
            key[i] = (ok > key[i]) ? ok : key[i];
        }
        if (lm == 0) {
            const int row = mtile * 16 + 8 * half + i;
            atomicAdd(&rsS[row], rs[i]);                  // ds_add_f32
            atomicMax(&keyS[row], key[i]);                // ds_max_u64
        }
    }
    __syncthreads();

    // ---------------- phase 2: normalize q and stream to HBM ----------------
    {
        v4f*       qO  = (v4f*)(q_out + (size_t)wg * BM * C_CODES);
        const v4f* qs4 = (const v4f*)qS;
        for (int row = 0; row < BM; ++row) {
            const float inv = 1.0f / rsS[row];
            v4f v = qs4[row * 256 + tid];
            v *= inv;
            __builtin_nontemporal_store(v, &qO[row * 256 + tid]);  // keep L2 for codebook
        }
    }

    // ---------------- phase 3: z_q gather + loss partial ----------------
    {
        const int row = tid >> 2;
        const int seg = tid & 3;
        const unsigned n = ~(unsigned)(keyS[row] & 0xFFFFFFFFull); // decode argmax col
        const v4f* crow = (const v4f*)(cb + (size_t)n * D_DIM + seg * 16);
        const v4f* zrow = (const v4f*)(zS + row * D_DIM + seg * 16);
        v4f* zqO = (v4f*)(zq_out + ((size_t)(wg * BM + row)) * D_DIM + seg * 16);
        float lsum = 0.0f;
        #pragma unroll
        for (int j = 0; j < 4; ++j) {
            v4f cv = crow[j];
            v4f zv = zrow[j];
            v4f dv = cv - zv;
            lsum += dv[0] * dv[0] + dv[1] * dv[1] + dv[2] * dv[2] + dv[3] * dv[3];
            __builtin_nontemporal_store(cv, &zqO[j]);     // z_q_st == z_q numerically
        }
        atomicAdd(lossS, lsum);
    }
    __syncthreads();
    if (tid == 0) loss_ws[wg] = lossS[0];
}

// finalize: loss = 1.25 * mean((z_q - z)^2)   (both stop-grad terms are equal)
__global__ void vq_loss_reduce(const float* __restrict__ ws, float* __restrict__ out)
{
    __shared__ float s[256];
    const int t = threadIdx.x;
    s[t] = ws[t] + ws[t + 256] + ws[t + 512] + ws[t + 768];
    __syncthreads();
    for (int off = 128; off > 0; off >>= 1) {
        if (t < off) s[t] += s[t + off];
        __syncthreads();
    }
    if (t == 0) out[0] = s[0] * (1.25f / ((float)B_ROWS * (float)D_DIM));
}

extern "C" void kernel_launch(void* const* d_in, const int* in_sizes, int n_in,
                              void* d_out, int out_size, void* d_ws, size_t ws_size,
                              hipStream_t stream)
{
    const float* z  = (const float*)d_in[0];   // [65536,64]
    const float* cb = (const float*)d_in[1];   // [1024,64]
    float* q    = (float*)d_out;                         // [65536,1024]
    float* zq   = q + (size_t)B_ROWS * C_CODES;          // [65536,64]
    float* loss = zq + (size_t)B_ROWS * D_DIM;           // [1]
    float* ws   = (float*)d_ws;                          // 1024 partials

    const size_t smem = (size_t)(BM * C_CODES + BM * D_DIM + BM + BM) * sizeof(float)
                      + (size_t)BM * sizeof(unsigned long long) + sizeof(float);

    vq_main<<<NBLK, 256, smem, stream>>>(z, cb, q, zq, ws);
    vq_loss_reduce<<<1, 256, 0, stream>>>(ws, loss);
}